// RuleLayer_18537078849734
// MI455X (gfx1250) — compile-verified
//
#include <hip/hip_runtime.h>

// out[b, i0..i5] = prod_d in[b, i_d, d], in shape (512, 8, 6) fp32, out 512 x 8^6.
//
// View out[b] as a 512x512 row-major matrix:
//   row r = i0*64 + i1*8 + i2   -> pA[r] = x0[i0]*x1[i1]*x2[i2]
//   col c = i3*64 + i4*8 + i5   -> pB[c] = x3[i3]*x4[i4]*x5[i5]
//   out[b][r][c] = pA[r] * pB[c]     (rank-1 outer product)
//
// Rank-1 fp32 outer product == V_WMMA_F32_16X16X4_F32 with only the K=0
// column of A / row of B populated: D = A(:,0) * B(0,:) + 0. Exact fp32.
//
// Store-bandwidth bound: 512 MiB / 23.3 TB/s ~ 23 us floor. Non-temporal
// stores keep the streaming output from thrashing the 192 MB L2.
//
// v2: removed the divergent (lane<16 ? lds : 0) loads that generated
// s_and_saveexec + branches in the inner loop. All lanes load (lanes 16-31
// read the same in-range address -> broadcast, finite value) and the K=0
// lane selection is done with a per-lane {1,0} float mask hoisted out of
// the loop. Inner loop is now branch-free: 2 ds_load + 2 v_mul + v_wmma +
// 8 clause'd NT stores.

typedef __attribute__((ext_vector_type(2))) float v2f;
typedef __attribute__((ext_vector_type(8))) float v8f;

#define ROWS 512
#define COLS 512
#define SLICES 8  // tile-row slices per batch; each workgroup: 4 tile-rows x 32 tile-cols

__global__ __launch_bounds__(256) void outer6_wmma_kernel(const float* __restrict__ in,
                                                          float* __restrict__ out) {
    __shared__ float sPA[ROWS];
    __shared__ float sPB[COLS];

    const int wg = blockIdx.x;       // 0 .. 512*SLICES-1
    const int b  = wg >> 3;          // batch
    const int s  = wg & (SLICES - 1);
    const int t  = threadIdx.x;      // 0..255

    const float* xb = in + b * 48;   // in[b, m, d] = xb[m*6 + d]

    // Build pA / pB in LDS (each thread fills 2 entries of each).
    for (int k = t; k < ROWS; k += 256) {
        const int a0 = (k >> 6) & 7, a1 = (k >> 3) & 7, a2 = k & 7;
        sPA[k] = xb[a0 * 6 + 0] * xb[a1 * 6 + 1] * xb[a2 * 6 + 2];
        sPB[k] = xb[a0 * 6 + 3] * xb[a1 * 6 + 4] * xb[a2 * 6 + 5];
    }
    __syncthreads();

    const int wave = t >> 5;                    // 0..7
    const int lane = t & 31;
    const int ln15 = lane & 15;
    const float kmask = (lane < 16) ? 1.0f : 0.0f;  // K=0 lives in lanes 0-15
    const int rowOff = (lane >> 4) << 3;        // 0 or 8

    float* outb = out + (size_t)b * (ROWS * (size_t)COLS);

    // This slice: 4 tile-rows x 32 tile-cols = 128 tiles of 16x16.
    for (int tile = wave; tile < 4 * 32; tile += 8) {
        const int trl = tile >> 5;   // 0..3  local tile-row
        const int tc  = tile & 31;   // 0..31 tile-col
        const int row_base = (s * 4 + trl) * 16;
        const int col_base = tc * 16;

        // A: 16x4 f32 (2 VGPRs). VGPR0 lanes0-15 = K=0 column = pA tile; rest 0.
        v2f a;
        a.x = sPA[row_base + ln15] * kmask;
        a.y = 0.0f;
        // B: 4x16 f32 (2 VGPRs). VGPR0 lanes0-15 = K=0 row = pB tile; rest 0.
        v2f bb;
        bb.x = sPB[col_base + ln15] * kmask;
        bb.y = 0.0f;

        v8f c = {};
        // 8 args: (neg_a, A, neg_b, B, c_mod, C, reuse_a, reuse_b)
        v8f d = __builtin_amdgcn_wmma_f32_16x16x4_f32(
            false, a, false, bb, (short)0, c, false, false);

        // D layout: VGPR j -> row M=j (lanes 0-15) / M=j+8 (lanes 16-31), N = lane&15.
        float* p = outb + (size_t)(row_base + rowOff) * COLS + (col_base + ln15);
#pragma unroll
        for (int j = 0; j < 8; ++j) {
            __builtin_nontemporal_store(d[j], p + (size_t)j * COLS);
        }
    }
}

extern "C" void kernel_launch(void* const* d_in, const int* in_sizes, int n_in,
                              void* d_out, int out_size, void* d_ws, size_t ws_size,
                              hipStream_t stream) {
    const float* in = (const float*)d_in[0];
    float* out      = (float*)d_out;
    dim3 grid(512 * SLICES);
    dim3 block(256);
    outer6_wmma_kernel<<<grid, block, 0, stream>>>(in, out);
}